// AgnosticNonlinearInteractionBlock_343597384378
// MI455X (gfx1250) — compile-verified
//
#include <hip/hip_runtime.h>
#include <hip/hip_bf16.h>

// Problem constants (match reference)
#define NN 10000
#define EE 160000
#define CC 128
#define AA 10
#define RR 8
#define HH 64

typedef float v2f __attribute__((ext_vector_type(2)));
typedef float v8f __attribute__((ext_vector_type(8)));

// D = A(16x4,f32) * B(4x16,f32) + C(16x16,f32)
// Fragment layouts per CDNA5 ISA 7.12.2:
//  A: lane L holds row M=L&15; VGPR j = K = k0 + 2*(L>>4) + j
//  B: VGPR j = row K = k0 + 2*(L>>4) + j, col N = L&15
//  C/D: VGPR r = element (M = r + 8*(L>>4), N = L&15)
#define WMMA_F32(c, a, b) \
  (c) = __builtin_amdgcn_wmma_f32_16x16x4_f32(false, (a), false, (b), (short)0, (c), false, false)

__device__ __forceinline__ float silu_f(float x) {
  return x / (1.0f + __expf(-x));
}

// ---------------------------------------------------------------------------
// zero workspace accumulators
__global__ void zero_kernel(float* __restrict__ p, long long count) {
  long long i = (long long)blockIdx.x * blockDim.x + threadIdx.x;
  long long stride = (long long)gridDim.x * blockDim.x;
  for (; i < count; i += stride) p[i] = 0.0f;
}

// ---------------------------------------------------------------------------
// Pack a row-major K x Nf weight matrix into per-lane WMMA B-fragment order:
// Wp[((nt*(K/4) + ks)*32 + lane)*2 + j] = W[(4*ks + 2*(lane>>4) + j)*Nf + nt*16 + (lane&15)]
// After packing, a wave's B fragment for (nt, ks) is one coalesced 256B b64 load.
__global__ void pack_kernel(const float* __restrict__ W, float* __restrict__ Wp,
                            int K, int Nf) {
  const int idx = blockIdx.x * blockDim.x + threadIdx.x;
  const int ksteps = K >> 2;
  const int total = (Nf >> 4) * ksteps * 32;
  if (idx >= total) return;
  const int lane = idx & 31;
  const int t = idx >> 5;
  const int ks = t % ksteps;
  const int nt = t / ksteps;
  const int k = ks * 4 + 2 * (lane >> 4);
  const int col = nt * 16 + (lane & 15);
  Wp[idx * 2 + 0] = W[k * Nf + col];
  Wp[idx * 2 + 1] = W[(k + 1) * Nf + col];
}

// ---------------------------------------------------------------------------
// Node up-projection: s = nf[:, :128] @ W_up0 / sqrt(128)
//                     v[n,d,i] = sum_c nf[n,128+3c+i] * W_up1[c,d] / sqrt(128)
// One wave per 16-node tile; B fragments from packed weights (b64, imm offsets).
__global__ void __launch_bounds__(256) up_kernel(
    const float* __restrict__ nf,
    const float* __restrict__ Pup0,   // packed 128x128
    const float* __restrict__ Pup1,   // packed 128x128
    float* __restrict__ s, float* __restrict__ v) {
  const int lane = threadIdx.x & 31;
  const int wid  = threadIdx.x >> 5;
  const int half = lane >> 4;
  const int rowi = lane & 15;
  const int tile = blockIdx.x * 8 + wid;
  if (tile >= NN / 16) return;               // whole-wave uniform exit
  const int n0 = tile * 16;
  const float inv = 0.08838834764831845f;    // 1/sqrt(128)

  // scalar channel: K=128 -> 32 ksteps, 8 n-tiles
  {
    const float* ab = nf + (size_t)(n0 + rowi) * 512 + 2 * half;  // 8B aligned
    for (int nt = 0; nt < 8; ++nt) {
      const v2f* bp = (const v2f*)Pup0 + (size_t)nt * 32 * 32 + lane;
      v8f c = {};
#pragma unroll
      for (int ks = 0; ks < 32; ++ks) {
        v2f a = *(const v2f*)(ab + 4 * ks);
        v2f b = bp[ks * 32];
        WMMA_F32(c, a, b);
      }
#pragma unroll
      for (int r = 0; r < 8; ++r)
        s[(size_t)(n0 + r + 8 * half) * 128 + nt * 16 + rowi] = c[r] * inv;
    }
  }
  // vector channels: A strided by 3 in K (two b32 loads, imm offsets)
  for (int i = 0; i < 3; ++i) {
    const float* ab = nf + (size_t)(n0 + rowi) * 512 + 128 + 6 * half + i;
    for (int nt = 0; nt < 8; ++nt) {
      const v2f* bp = (const v2f*)Pup1 + (size_t)nt * 32 * 32 + lane;
      v8f c = {};
#pragma unroll
      for (int ks = 0; ks < 32; ++ks) {
        v2f a;
        a.x = ab[12 * ks];
        a.y = ab[12 * ks + 3];
        v2f b = bp[ks * 32];
        WMMA_F32(c, a, b);
      }
#pragma unroll
      for (int r = 0; r < 8; ++r)
        v[(size_t)(n0 + r + 8 * half) * 384 + (nt * 16 + rowi) * 3 + i] = c[r] * inv;
    }
  }
}

// ---------------------------------------------------------------------------
// Fused edge pipeline: 4-layer MLP (WMMA f32) -> tensor-product messages ->
// native f32 atomic segment-sum into M_s / M_v. One wave per 16-edge tile;
// activations live in a per-wave 16x64 LDS tile (same-wave DS ops in-order).
__global__ void __launch_bounds__(256) edge_kernel(
    const float* __restrict__ ef,
    const float* __restrict__ ea,
    const int* __restrict__ eidx,
    const float* __restrict__ P1,   // packed 8x64
    const float* __restrict__ P2,   // packed 64x64
    const float* __restrict__ P3,   // packed 64x64
    const float* __restrict__ P4,   // packed 64x512
    const float* __restrict__ s,
    const float* __restrict__ v,
    float* __restrict__ Ms, float* __restrict__ Mv) {
  __shared__ float hbuf[8][16 * 64];         // per-wave activation tile, 32 KB
  const int lane = threadIdx.x & 31;
  const int wid  = threadIdx.x >> 5;
  const int half = lane >> 4;
  const int rowi = lane & 15;
  const int tile = blockIdx.x * 8 + wid;     // grid sized exactly
  const int eb   = tile * 16;
  float* hb = hbuf[wid];
  const float* hab = hb + rowi * 64 + 2 * half;  // A-frag base in LDS (8B aligned)

  v8f acc[4];
  // ---- layer 1: (16x8) @ (8x64), /sqrt(8), silu  [ksteps=2]
  {
    const float* ab = ef + (size_t)(eb + rowi) * RR + 2 * half;
    for (int nt = 0; nt < 4; ++nt) {
      const v2f* bp = (const v2f*)P1 + (size_t)nt * 2 * 32 + lane;
      v8f c = {};
#pragma unroll
      for (int ks = 0; ks < 2; ++ks) {
        v2f a = *(const v2f*)(ab + 4 * ks);
        v2f b = bp[ks * 32];
        WMMA_F32(c, a, b);
      }
      acc[nt] = c;
    }
  }
  for (int nt = 0; nt < 4; ++nt)
#pragma unroll
    for (int r = 0; r < 8; ++r)
      hb[(r + 8 * half) * 64 + nt * 16 + rowi] = silu_f(acc[nt][r] * 0.3535533905932738f);

  // ---- layers 2 & 3: (16x64) @ (64x64), /8, silu  [ksteps=16]
  for (int layer = 0; layer < 2; ++layer) {
    const float* P = layer ? P3 : P2;
    for (int nt = 0; nt < 4; ++nt) {
      const v2f* bp = (const v2f*)P + (size_t)nt * 16 * 32 + lane;
      v8f c = {};
#pragma unroll
      for (int ks = 0; ks < 16; ++ks) {
        v2f a = *(const v2f*)(hab + 4 * ks);
        v2f b = bp[ks * 32];
        WMMA_F32(c, a, b);
      }
      acc[nt] = c;
    }
    for (int nt = 0; nt < 4; ++nt)
#pragma unroll
      for (int r = 0; r < 8; ++r)
        hb[(r + 8 * half) * 64 + nt * 16 + rowi] = silu_f(acc[nt][r] * 0.125f);
  }

  // per-edge metadata: D-VGPR r of this lane maps to edge row (r + 8*half)
  int snd[8], rcv[8];
  float y0e[8], y1e[8][3];
#pragma unroll
  for (int r = 0; r < 8; ++r) {
    const int e = eb + r + 8 * half;
    snd[r] = eidx[e];
    rcv[r] = eidx[EE + e];
    y0e[r]    = ea[e * 4 + 0];
    y1e[r][0] = ea[e * 4 + 1];
    y1e[r][1] = ea[e * 4 + 2];
    y1e[r][2] = ea[e * 4 + 3];
  }

  // ---- layer 4: (16x64) @ (64x512), /8 -> tp_w tile, fused into messages.
  // nt>>3 selects weight block: 0=ws1, 1=ws2, 2=wv1, 3=wv2
  for (int nt = 0; nt < 32; ++nt) {
    const v2f* bp = (const v2f*)P4 + (size_t)nt * 16 * 32 + lane;
    v8f c = {};
#pragma unroll
    for (int ks = 0; ks < 16; ++ks) {
      v2f a = *(const v2f*)(hab + 4 * ks);
      v2f b = bp[ks * 32];
      WMMA_F32(c, a, b);
    }
    const int blk = nt >> 3;
    const int ch  = (nt & 7) * 16 + rowi;    // channel 0..127
#pragma unroll
    for (int r = 0; r < 8; ++r) {
      const float w  = c[r] * 0.125f;        // /sqrt(H)
      const int   sn = snd[r];
      const int   rc = rcv[r];
      if (blk == 0) {                        // m_s1 = ws1 * s_j * y0
        unsafeAtomicAdd(&Ms[(size_t)rc * 256 + ch], w * s[(size_t)sn * 128 + ch] * y0e[r]);
      } else if (blk == 1) {                 // m_s2 = ws2 * <v_j, y1> / sqrt(3)
        const float* vp = &v[(size_t)sn * 384 + ch * 3];
        const float dot = vp[0] * y1e[r][0] + vp[1] * y1e[r][1] + vp[2] * y1e[r][2];
        unsafeAtomicAdd(&Ms[(size_t)rc * 256 + 128 + ch], w * dot * 0.5773502691896258f);
      } else if (blk == 2) {                 // m_v1 = wv1 * s_j * y1[i]
        const float base = w * s[(size_t)sn * 128 + ch];
        float* mp = &Mv[(size_t)rc * 768 + ch * 3];
        unsafeAtomicAdd(mp + 0, base * y1e[r][0]);
        unsafeAtomicAdd(mp + 1, base * y1e[r][1]);
        unsafeAtomicAdd(mp + 2, base * y1e[r][2]);
      } else {                               // m_v2 = wv2 * v_j * y0
        const float* vp = &v[(size_t)sn * 384 + ch * 3];
        const float wy  = w * y0e[r];
        float* mp = &Mv[(size_t)rc * 768 + (128 + ch) * 3];
        unsafeAtomicAdd(mp + 0, wy * vp[0]);
        unsafeAtomicAdd(mp + 1, wy * vp[1]);
        unsafeAtomicAdd(mp + 2, wy * vp[2]);
      }
    }
  }
}

// ---------------------------------------------------------------------------
// Output linears: out_s = M_s @ W_lin0 / 256 ; out_v[:,:,i] = M_v[:,:,i] @ W_lin1 / 256
// Stored in final layout: osv[n][0:128]=out_s, osv[n][128+3d+i]=out_v[n,d,i]
__global__ void __launch_bounds__(256) lin_kernel(
    const float* __restrict__ Ms,
    const float* __restrict__ Mv,
    const float* __restrict__ Pl0,   // packed 256x128
    const float* __restrict__ Pl1,   // packed 256x128
    float* __restrict__ osv) {
  const int lane = threadIdx.x & 31;
  const int wid  = threadIdx.x >> 5;
  const int half = lane >> 4;
  const int rowi = lane & 15;
  const int tile = blockIdx.x * 8 + wid;
  if (tile >= NN / 16) return;
  const int n0 = tile * 16;
  const float inv = 1.0f / 256.0f;           // 1/(sqrt(2C)*AVG_NEI)

  // out_s: K=256 -> 64 ksteps
  {
    const float* ab = Ms + (size_t)(n0 + rowi) * 256 + 2 * half;
    for (int nt = 0; nt < 8; ++nt) {
      const v2f* bp = (const v2f*)Pl0 + (size_t)nt * 64 * 32 + lane;
      v8f c = {};
#pragma unroll
      for (int ks = 0; ks < 64; ++ks) {
        v2f a = *(const v2f*)(ab + 4 * ks);
        v2f b = bp[ks * 32];
        WMMA_F32(c, a, b);
      }
#pragma unroll
      for (int r = 0; r < 8; ++r)
        osv[(size_t)(n0 + r + 8 * half) * 512 + nt * 16 + rowi] = c[r] * inv;
    }
  }
  // out_v: A strided by 3 in K
  for (int i = 0; i < 3; ++i) {
    const float* ab = Mv + (size_t)(n0 + rowi) * 768 + 6 * half + i;
    for (int nt = 0; nt < 8; ++nt) {
      const v2f* bp = (const v2f*)Pl1 + (size_t)nt * 64 * 32 + lane;
      v8f c = {};
#pragma unroll
      for (int ks = 0; ks < 64; ++ks) {
        v2f a;
        a.x = ab[12 * ks];
        a.y = ab[12 * ks + 3];
        v2f b = bp[ks * 32];
        WMMA_F32(c, a, b);
      }
#pragma unroll
      for (int r = 0; r < 8; ++r)
        osv[(size_t)(n0 + r + 8 * half) * 512 + 128 + 3 * (nt * 16 + rowi) + i] = c[r] * inv;
    }
  }
}

// ---------------------------------------------------------------------------
// Attr-gated skip: o[n,d] = sum_c out[n,c] * W_skip[c, attr(n), d] / sqrt(C*A).
// One-hot attr selection is random per node -> VALU FMA (1.3 GFLOP, negligible).
__global__ void __launch_bounds__(128) skip_kernel(
    const float* __restrict__ na,
    const float* __restrict__ osv,
    const float* __restrict__ Wsk0,
    const float* __restrict__ Wsk1,
    float* __restrict__ out) {
  __shared__ float row[512];
  __shared__ int attr_s;
  const int n = blockIdx.x;
  const int d = threadIdx.x;                 // 0..127
  for (int t = d; t < 512; t += 128) row[t] = osv[(size_t)n * 512 + t];
  if (d == 0) {
    int a = 0;
    for (int q = 0; q < AA; ++q)
      if (na[n * AA + q] > 0.5f) a = q;
    attr_s = a;
  }
  __syncthreads();
  const int a = attr_s;
  const float* w0p = Wsk0 + (size_t)a * 128 + d;
  const float* w1p = Wsk1 + (size_t)a * 128 + d;
  float a0 = 0.f, a1 = 0.f, a2 = 0.f, a3 = 0.f;
  for (int cth = 0; cth < 128; ++cth) {
    const float w0 = w0p[(size_t)cth * AA * 128];
    const float w1 = w1p[(size_t)cth * AA * 128];
    a0 = fmaf(row[cth], w0, a0);
    a1 = fmaf(row[128 + 3 * cth + 0], w1, a1);
    a2 = fmaf(row[128 + 3 * cth + 1], w1, a2);
    a3 = fmaf(row[128 + 3 * cth + 2], w1, a3);
  }
  const float invfan = 0.02795084971874737f; // 1/sqrt(128*10)
  out[(size_t)n * 512 + d] = a0 * invfan;
  out[(size_t)n * 512 + 128 + 3 * d + 0] = a1 * invfan;
  out[(size_t)n * 512 + 128 + 3 * d + 1] = a2 * invfan;
  out[(size_t)n * 512 + 128 + 3 * d + 2] = a3 * invfan;
}

// ---------------------------------------------------------------------------
extern "C" void kernel_launch(void* const* d_in, const int* in_sizes, int n_in,
                              void* d_out, int out_size, void* d_ws, size_t ws_size,
                              hipStream_t stream) {
  const float* node_attrs = (const float*)d_in[0];
  const float* node_feats = (const float*)d_in[1];
  const float* edge_attrs = (const float*)d_in[2];
  const float* edge_feats = (const float*)d_in[3];
  const int*   edge_index = (const int*)d_in[4];
  const float* W_up0  = (const float*)d_in[5];
  const float* W_up1  = (const float*)d_in[6];
  const float* W_mlp1 = (const float*)d_in[7];
  const float* W_mlp2 = (const float*)d_in[8];
  const float* W_mlp3 = (const float*)d_in[9];
  const float* W_mlp4 = (const float*)d_in[10];
  const float* W_lin0 = (const float*)d_in[11];
  const float* W_lin1 = (const float*)d_in[12];
  const float* W_skip0 = (const float*)d_in[13];
  const float* W_skip1 = (const float*)d_in[14];

  // workspace layout (floats)
  float* ws  = (float*)d_ws;
  float* s   = ws;                            // N*128
  float* v   = s + (size_t)NN * 128;          // N*384
  float* Ms  = v + (size_t)NN * 384;          // N*256  (zeroed)
  float* Mv  = Ms + (size_t)NN * 256;         // N*768  (zeroed, contiguous with Ms)
  float* osv = Mv + (size_t)NN * 768;         // N*512
  // packed weights (B-fragment order)
  float* Pm1 = osv + (size_t)NN * 512;        // 512
  float* Pm2 = Pm1 + 512;                     // 4096
  float* Pm3 = Pm2 + 4096;                    // 4096
  float* Pm4 = Pm3 + 4096;                    // 32768
  float* Pu0 = Pm4 + 32768;                   // 16384
  float* Pu1 = Pu0 + 16384;                   // 16384
  float* Pl0 = Pu1 + 16384;                   // 32768
  float* Pl1 = Pl0 + 32768;                   // 32768

  auto pack = [&](const float* W, float* Wp, int K, int Nf) {
    const int total = (Nf / 16) * (K / 4) * 32;
    pack_kernel<<<(total + 255) / 256, 256, 0, stream>>>(W, Wp, K, Nf);
  };
  pack(W_mlp1, Pm1, 8, 64);
  pack(W_mlp2, Pm2, 64, 64);
  pack(W_mlp3, Pm3, 64, 64);
  pack(W_mlp4, Pm4, 64, 512);
  pack(W_up0,  Pu0, 128, 128);
  pack(W_up1,  Pu1, 128, 128);
  pack(W_lin0, Pl0, 256, 128);
  pack(W_lin1, Pl1, 256, 128);

  zero_kernel<<<2048, 256, 0, stream>>>(Ms, (long long)NN * 1024);
  up_kernel<<<(NN / 16 + 7) / 8, 256, 0, stream>>>(node_feats, Pu0, Pu1, s, v);
  edge_kernel<<<EE / 16 / 8, 256, 0, stream>>>(edge_feats, edge_attrs, edge_index,
                                               Pm1, Pm2, Pm3, Pm4, s, v, Ms, Mv);
  lin_kernel<<<(NN / 16 + 7) / 8, 256, 0, stream>>>(Ms, Mv, Pl0, Pl1, osv);
  skip_kernel<<<NN, 128, 0, stream>>>(node_attrs, osv, W_skip0, W_skip1, (float*)d_out);
}